// GIN_4681514352775
// MI455X (gfx1250) — compile-verified
//
#include <hip/hip_runtime.h>
#include <hip/hip_bf16.h>

// ---------------------------------------------------------------------------
// GIN forward for MI455X (gfx1250, wave32). f32 WMMA 16x16x4 for the MLPs.
// Weights staged once per block into LDS (pre-swizzled for ds_load_b64).
// ---------------------------------------------------------------------------

#define N_NODES   100000
#define N_EDGES   1200000
#define DCH       64
#define N_GRAPHS  512
#define N_CLASSES 10
#define BN_EPS    1e-5f
#define NSTRIPS   (N_NODES / 16)      // 6250 exactly

typedef float v2f __attribute__((ext_vector_type(2)));
typedef float v8f __attribute__((ext_vector_type(8)));

__device__ __forceinline__ v8f wmma_f32(v2f a, v2f b, v8f c) {
  // 8 args: (neg_a, A, neg_b, B, c_mod, C, reuse_a, reuse_b)
  return __builtin_amdgcn_wmma_f32_16x16x4_f32(false, a, false, b, (short)0, c,
                                               false, false);
}

// -------------------------- utility kernels --------------------------------

__global__ void kcopy4(float4* __restrict__ dst, const float4* __restrict__ src,
                       int n4) {
  int i = blockIdx.x * blockDim.x + threadIdx.x;
  int stride = gridDim.x * blockDim.x;
  for (; i < n4; i += stride) dst[i] = src[i];
}

__global__ void kzero(float* __restrict__ p, int n) {
  int i = blockIdx.x * blockDim.x + threadIdx.x;
  int stride = gridDim.x * blockDim.x;
  for (; i < n; i += stride) p[i] = 0.0f;
}

// ------------------------- edge scatter-add --------------------------------
// agg[dst] += h[src]; 16 lanes per edge, float4 gather + 4 f32 atomics/lane.
// All of h (25.6 MB) is L2-resident; this phase is L2-atomic bound.

__global__ void kedge(const float* __restrict__ h, float* __restrict__ agg,
                      const int* __restrict__ ei, int nE) {
  int t = blockIdx.x * blockDim.x + threadIdx.x;
  int e = t >> 4;
  if (e >= nE) return;
  int c4 = (t & 15) * 4;
  int src = ei[e];
  int dst = ei[nE + e];
  float4 v = *(const float4*)&h[src * DCH + c4];
  float* o = &agg[dst * DCH + c4];
  atomicAdd(o + 0, v.x);
  atomicAdd(o + 1, v.y);
  atomicAdd(o + 2, v.z);
  atomicAdd(o + 3, v.w);
}

// ------------------- fused 2-GEMM MLP via f32 WMMA -------------------------
// z (in/out, N_NODES x 64):  z <- relu(z@W1 + b1) @ W2 + b2   (in place)
// One wave per 16-node strip; 4 waves per block. W1/W2 staged once per block
// into LDS, swizzled so the (W[k][n], W[k+1][n]) B-fragment is one b64 load.

__global__ __launch_bounds__(128) void kmlp(float* __restrict__ z,
                                            const float* __restrict__ W1,
                                            const float* __restrict__ b1,
                                            const float* __restrict__ W2,
                                            const float* __restrict__ b2,
                                            int nstrips) {
  __shared__ float sW1[4096];      // swizzled: [(k>>1)*128 + n*2 + (k&1)]
  __shared__ float sW2[4096];
  __shared__ float ldsZ1[4 * 1024];  // per-wave z1 strip (16x64)

  const int lane = threadIdx.x & 31;
  const int wave = threadIdx.x >> 5;

  // Cooperative, swizzled weight staging (float4 global reads).
  for (int i4 = threadIdx.x; i4 < 1024; i4 += 128) {
    int k = i4 >> 4;             // row 0..63
    int n = (i4 & 15) * 4;       // col
    int base = (k >> 1) * 128 + n * 2 + (k & 1);
    float4 w = ((const float4*)W1)[i4];
    sW1[base + 0] = w.x; sW1[base + 2] = w.y;
    sW1[base + 4] = w.z; sW1[base + 6] = w.w;
    w = ((const float4*)W2)[i4];
    sW2[base + 0] = w.x; sW2[base + 2] = w.y;
    sW2[base + 4] = w.z; sW2[base + 6] = w.w;
  }
  __syncthreads();

  const int strip = blockIdx.x * 4 + wave;
  if (strip >= nstrips) return;              // wave-uniform: EXEC stays full
  float* myZ1 = &ldsZ1[wave * 1024];
  const int m0 = strip * 16;

  const int mr = lane & 15;           // A row / B,C column within tile
  const int kh = (lane >> 4) * 2;     // lanes 16-31 hold K+2,K+3
  const int rbase = (lane >> 4) * 8;  // C/D rows: r (lanes 0-15), r+8 (16-31)

  // ---- GEMM1: z1 = relu(A @ W1 + b1), A straight from global ----
  const float* zrow = &z[(m0 + mr) * DCH];
  v8f acc[4] = {};
#pragma unroll
  for (int kk = 0; kk < 16; ++kk) {
    const int kb = kk * 4 + kh;                      // even
    v2f a = *(const v2f*)&zrow[kb];
    const float* wrow = &sW1[(kb >> 1) * 128];
#pragma unroll
    for (int t = 0; t < 4; ++t) {
      v2f b = *(const v2f*)&wrow[(t * 16 + mr) * 2];
      acc[t] = wmma_f32(a, b, acc[t]);
    }
  }
  // bias + relu, write z1 to this wave's LDS strip (row-major 16x64)
#pragma unroll
  for (int t = 0; t < 4; ++t) {
    const float bias = b1[t * 16 + mr];
#pragma unroll
    for (int r = 0; r < 8; ++r) {
      float v = acc[t][r] + bias;
      myZ1[(rbase + r) * DCH + t * 16 + mr] = v > 0.0f ? v : 0.0f;
    }
  }

  // ---- GEMM2: z2 = z1 @ W2 + b2 ---- (same-wave LDS is in order)
  v8f acc2[4] = {};
#pragma unroll
  for (int kk = 0; kk < 16; ++kk) {
    const int kb = kk * 4 + kh;
    v2f a = *(const v2f*)&myZ1[mr * DCH + kb];
    const float* wrow = &sW2[(kb >> 1) * 128];
#pragma unroll
    for (int t = 0; t < 4; ++t) {
      v2f b = *(const v2f*)&wrow[(t * 16 + mr) * 2];
      acc2[t] = wmma_f32(a, b, acc2[t]);
    }
  }
  // bias, store z2 back to global (in place; stores data-depend on all A loads)
#pragma unroll
  for (int t = 0; t < 4; ++t) {
    const float bias = b2[t * 16 + mr];
#pragma unroll
    for (int r = 0; r < 8; ++r) {
      z[(m0 + rbase + r) * DCH + t * 16 + mr] = acc2[t][r] + bias;
    }
  }
}

// ------------------------- BatchNorm statistics ----------------------------
// stats[c] = sum over rows, stats[64+c] = sum of squares.

__global__ void kstats(const float* __restrict__ z, float* __restrict__ stats,
                       int nrows) {
  __shared__ float ls[256], ls2[256];
  const int c = threadIdx.x & 63;
  const int rg = threadIdx.x >> 6;  // 0..3
  float s = 0.0f, s2 = 0.0f;
  for (int r = blockIdx.x * 4 + rg; r < nrows; r += gridDim.x * 4) {
    float v = z[r * DCH + c];
    s += v;
    s2 += v * v;
  }
  ls[threadIdx.x] = s;
  ls2[threadIdx.x] = s2;
  __syncthreads();
  if (rg == 0) {
    s = ls[c] + ls[64 + c] + ls[128 + c] + ls[192 + c];
    s2 = ls2[c] + ls2[64 + c] + ls2[128 + c] + ls2[192 + c];
    atomicAdd(&stats[c], s);
    atomicAdd(&stats[64 + c], s2);
  }
}

__global__ void kbnrelu4(const float4* __restrict__ z4, float4* __restrict__ h4,
                         const float* __restrict__ stats,
                         const float* __restrict__ g,
                         const float* __restrict__ b, int n4, float invN) {
  int i = blockIdx.x * blockDim.x + threadIdx.x;
  int stride = gridDim.x * blockDim.x;
  for (; i < n4; i += stride) {
    int c4 = (i & 15) * 4;
    float4 v = z4[i];
    float4 o;
    const float* vp = &v.x;
    float* op = &o.x;
#pragma unroll
    for (int j = 0; j < 4; ++j) {
      int c = c4 + j;
      float mu = stats[c] * invN;
      float var = stats[64 + c] * invN - mu * mu;
      float sc = g[c] * rsqrtf(var + BN_EPS);
      float u = (vp[j] - mu) * sc + b[c];
      op[j] = u > 0.0f ? u : 0.0f;
    }
    h4[i] = o;
  }
}

// --------------------------- global mean pool ------------------------------

__global__ void kpool4(const float4* __restrict__ h4,
                       const int* __restrict__ batch,
                       float* __restrict__ pooled, float* __restrict__ counts,
                       int n4) {
  int i = blockIdx.x * blockDim.x + threadIdx.x;
  int stride = gridDim.x * blockDim.x;
  for (; i < n4; i += stride) {
    int node = i >> 4;
    int c4 = (i & 15) * 4;
    int bg = batch[node];
    float4 v = h4[i];
    float* o = &pooled[bg * DCH + c4];
    atomicAdd(o + 0, v.x);
    atomicAdd(o + 1, v.y);
    atomicAdd(o + 2, v.z);
    atomicAdd(o + 3, v.w);
    if (c4 == 0) atomicAdd(&counts[bg], 1.0f);
  }
}

// ------------------------------- head --------------------------------------

__global__ void khead1(const float* __restrict__ pooled,
                       const float* __restrict__ counts,
                       const float* __restrict__ W,
                       const float* __restrict__ bias,
                       float* __restrict__ out) {
  int gr = blockIdx.x;        // 512
  int c = threadIdx.x;        // 64
  float inv = 1.0f / fmaxf(counts[gr], 1.0f);
  float acc = bias[c];
  for (int k = 0; k < 64; ++k) acc += pooled[gr * DCH + k] * inv * W[k * 64 + c];
  out[gr * 64 + c] = acc;
}

__global__ void kbn_small(float* __restrict__ x, const float* __restrict__ g,
                          const float* __restrict__ b, int rows, int cols) {
  int c = threadIdx.x;  // cols threads, single block
  float s = 0.0f, s2 = 0.0f;
  for (int r = 0; r < rows; ++r) {
    float v = x[r * cols + c];
    s += v;
    s2 += v * v;
  }
  float mu = s / (float)rows;
  float var = s2 / (float)rows - mu * mu;
  float sc = g[c] * rsqrtf(var + BN_EPS);
  float bb = b[c];
  for (int r = 0; r < rows; ++r) {
    float v = (x[r * cols + c] - mu) * sc + bb;
    x[r * cols + c] = v > 0.0f ? v : 0.0f;
  }
}

__global__ void khead2(const float* __restrict__ in, const float* __restrict__ W,
                       const float* __restrict__ bias, float* __restrict__ out) {
  int gr = blockIdx.x;  // 512
  int c = threadIdx.x;  // 32
  float acc = bias[c];
  for (int k = 0; k < 64; ++k) acc += in[gr * 64 + k] * W[k * 32 + c];
  out[gr * 32 + c] = acc;
}

__global__ void khead3(const float* __restrict__ in, const float* __restrict__ W,
                       const float* __restrict__ bias, float* __restrict__ out) {
  int gr = blockIdx.x;  // 512
  int c = threadIdx.x;  // 16 threads, 10 used
  if (c >= N_CLASSES) return;
  float acc = bias[c];
  for (int k = 0; k < 32; ++k) acc += in[gr * 32 + k] * W[k * N_CLASSES + c];
  out[gr * N_CLASSES + c] = acc;
}

// ------------------------------ launcher -----------------------------------

extern "C" void kernel_launch(void* const* d_in, const int* in_sizes, int n_in,
                              void* d_out, int out_size, void* d_ws,
                              size_t ws_size, hipStream_t stream) {
  (void)in_sizes; (void)n_in; (void)out_size; (void)ws_size;
  const float* x      = (const float*)d_in[0];
  const int*   ei     = (const int*)d_in[1];
  const int*   batch  = (const int*)d_in[2];
  const float* mlp_w1 = (const float*)d_in[3];
  const float* mlp_b1 = (const float*)d_in[4];
  const float* mlp_w2 = (const float*)d_in[5];
  const float* mlp_b2 = (const float*)d_in[6];
  const float* bn_g   = (const float*)d_in[7];
  const float* bn_b   = (const float*)d_in[8];
  const float* hw1    = (const float*)d_in[9];
  const float* hb1    = (const float*)d_in[10];
  const float* hg1    = (const float*)d_in[11];
  const float* hbb1   = (const float*)d_in[12];
  const float* hw2    = (const float*)d_in[13];
  const float* hb2    = (const float*)d_in[14];
  const float* hg2    = (const float*)d_in[15];
  const float* hbb2   = (const float*)d_in[16];
  const float* ow     = (const float*)d_in[17];
  const float* ob     = (const float*)d_in[18];
  float* out = (float*)d_out;

  float* ws     = (float*)d_ws;
  float* bufA   = ws;                     // h       : 6,400,000 f32
  float* bufZ   = bufA + N_NODES * DCH;   // z/agg   : 6,400,000 f32
  float* stats  = bufZ + N_NODES * DCH;   // 128 f32
  float* pooled = stats + 128;            // 512*64
  float* counts = pooled + N_GRAPHS * DCH;  // 512 (contiguous after pooled)
  float* tmp1   = counts + N_GRAPHS;      // 512*64
  float* tmp2   = tmp1 + N_GRAPHS * DCH;  // 512*32

  const int n4 = (N_NODES * DCH) / 4;     // 1,600,000 float4
  const float invN = 1.0f / (float)N_NODES;

  // h = x
  kcopy4<<<2048, 256, 0, stream>>>((float4*)bufA, (const float4*)x, n4);

  for (int l = 0; l < 3; ++l) {
    // z_pre = h  (then scatter adds neighbors -> h + sum_j h_j)
    kcopy4<<<2048, 256, 0, stream>>>((float4*)bufZ, (const float4*)bufA, n4);
    kedge<<<(N_EDGES * 16 + 255) / 256, 256, 0, stream>>>(bufA, bufZ, ei,
                                                          N_EDGES);
    // z = relu(z@W1+b1)@W2+b2  (WMMA, in place)
    kmlp<<<(NSTRIPS + 3) / 4, 128, 0, stream>>>(
        bufZ, mlp_w1 + l * DCH * DCH, mlp_b1 + l * DCH,
        mlp_w2 + l * DCH * DCH, mlp_b2 + l * DCH, NSTRIPS);
    // BN stats + normalize + relu -> h
    kzero<<<1, 128, 0, stream>>>(stats, 128);
    kstats<<<256, 256, 0, stream>>>(bufZ, stats, N_NODES);
    kbnrelu4<<<2048, 256, 0, stream>>>((const float4*)bufZ, (float4*)bufA,
                                       stats, bn_g + l * DCH, bn_b + l * DCH,
                                       n4, invN);
  }

  // global mean pool
  kzero<<<(N_GRAPHS * DCH + N_GRAPHS + 255) / 256, 256, 0, stream>>>(
      pooled, N_GRAPHS * DCH + N_GRAPHS);
  kpool4<<<2048, 256, 0, stream>>>((const float4*)bufA, batch, pooled, counts,
                                   n4);

  // head
  khead1<<<N_GRAPHS, 64, 0, stream>>>(pooled, counts, hw1, hb1, tmp1);
  kbn_small<<<1, 64, 0, stream>>>(tmp1, hg1, hbb1, N_GRAPHS, 64);
  khead2<<<N_GRAPHS, 32, 0, stream>>>(tmp1, hw2, hb2, tmp2);
  kbn_small<<<1, 32, 0, stream>>>(tmp2, hg2, hbb2, N_GRAPHS, 32);
  khead3<<<N_GRAPHS, 16, 0, stream>>>(tmp2, ow, ob, out);
}